// M_transform_66675072303670
// MI455X (gfx1250) — compile-verified
//
#include <hip/hip_runtime.h>
#include <hip/hip_bf16.h>
#include <math.h>

// Problem constants (from the reference)
#define TT    128                 // timesteps
#define LENM  32                  // band width
#define CCOLS (8192 * 64)         // N*F = 524288 columns
#define NTILE 8                   // T / 16 output row-tiles
#define NKK   12                  // 48-row K-span / 4 per WMMA

typedef __attribute__((ext_vector_type(2))) float v2f;
typedef __attribute__((ext_vector_type(4))) float v4f;
typedef __attribute__((ext_vector_type(8))) float v8f;

// ---------------------------------------------------------------------------
// Kernel 1: build softmax band weights directly in WMMA A-fragment layout.
// ws layout: ws[((tt*12 + kk)*32 + lane)*2 + q] = W[t][K] with
//   t = 16*tt + (lane & 15),  K = 4*kk + 2*(lane>>4) + q,
//   X-row j = jstart(tt) + K, jstart(tt) = max(0, 16*tt - 32).
// Zero outside the causal window -> uniform 12-step K loop in the main kernel.
// ---------------------------------------------------------------------------
__global__ void band_weights_66675072303670_kernel(
    const float* __restrict__ M_pad,   // [32][32]
    const float* __restrict__ M_big,   // [95][32]
    float* __restrict__ wsA)           // [8][12][32][2]
{
  const int tt    = blockIdx.x;        // 0..7
  const int lane  = threadIdx.x;       // 0..31
  const int M     = lane & 15;
  const int khalf = lane >> 4;
  const int t     = tt * 16 + M;                 // output row 0..127
  const int s     = (t < LENM) ? 0 : (t - LENM + 1);   // window start j
  const int L     = (t < LENM) ? (t + 1) : LENM;       // window length
  const int jstart = (tt < 2) ? 0 : (tt * 16 - 32);

  // logits row for this t (matches reference: t<32 -> M_pad[t];
  // t==32 reuses M_pad[31]; t>32 -> M_big[t-33])
  const float* logits;
  if (t < LENM)       logits = M_pad + t * LENM;
  else if (t == LENM) logits = M_pad + (LENM - 1) * LENM;
  else                logits = M_big + (t - LENM - 1) * LENM;

  float mx = -3.402823466e38f;
  for (int i = 0; i < L; ++i) mx = fmaxf(mx, logits[i]);
  float denom = 0.0f;
  for (int i = 0; i < L; ++i) denom += __expf(logits[i] - mx);
  const float rden = 1.0f / denom;

  for (int kk = 0; kk < NKK; ++kk) {
    float v0 = 0.0f, v1 = 0.0f;
    {
      const int K = kk * 4 + khalf * 2 + 0;
      const int j = jstart + K;
      if (j >= s && j <= t) v0 = __expf(logits[j - s] - mx) * rden;
    }
    {
      const int K = kk * 4 + khalf * 2 + 1;
      const int j = jstart + K;
      if (j >= s && j <= t) v1 = __expf(logits[j - s] - mx) * rden;
    }
    v2f val; val[0] = v0; val[1] = v1;
    *(v2f*)(wsA + ((size_t)(tt * NKK + kk) * 32 + lane) * 2) = val;
  }
}

// ---------------------------------------------------------------------------
// Kernel 2: banded mix via V_WMMA_F32_16X16X4_F32.
// Block: 256 threads (8 wave32s), owns a 128-column slice of X/out.
// Stage X[:,slice] once into LDS (coalesced b128, non-temporal), then each
// wave computes 16 columns x all 8 t-tiles with 12 K-step WMMAs per tile
// (two accumulator chains to break the WMMA RAW dependency).
//
// LDS pitch 136 words: rows j and j+2 (the two lane-halves of a B fragment)
// land 2*136 mod 64 = 16 banks apart -> ds_load_2addr is bank-conflict-free.
// ---------------------------------------------------------------------------
#define LDSLD 136   // LDS row pitch in floats (128*136*4 = 68 KB)

__global__ __launch_bounds__(256) void band_mix_66675072303670_kernel(
    const float* __restrict__ X,     // [128][CCOLS]
    const float* __restrict__ wsA,   // [8][12][32][2]
    float* __restrict__ out)         // [128][CCOLS]
{
  __shared__ float lds[TT * LDSLD];

  const int c0   = blockIdx.x * 128;
  const int tid  = threadIdx.x;
  const int lane = tid & 31;
  const int wave = tid >> 5;

  // ---- stage X slice: wave w loads rows 16w..16w+15, 512B per row ----
  // Streaming data, no cross-block reuse: non-temporal to keep L2 clean.
  #pragma unroll 4
  for (int r = 0; r < 16; ++r) {
    const int j = wave * 16 + r;
    const v4f v = __builtin_nontemporal_load(
        (const v4f*)(X + (size_t)j * CCOLS + c0 + lane * 4));
    *(v4f*)(&lds[j * LDSLD + lane * 4]) = v;
  }
  __syncthreads();

  const int cw    = wave * 16;     // wave's column base inside the slice
  const int M     = lane & 15;     // A-row / B-col / D-col index
  const int khalf = lane >> 4;

  for (int tt = 0; tt < NTILE; ++tt) {
    const int jstart = (tt < 2) ? 0 : (tt * 16 - 32);
    v8f acc0 = {};
    v8f acc1 = {};
    #pragma unroll
    for (int kk = 0; kk < NKK; ++kk) {
      // A fragment (weights): broadcast, L2-resident (24 KB total)
      const v2f a = *(const v2f*)(wsA + ((size_t)(tt * NKK + kk) * 32 + lane) * 2);
      // B fragment (data) from LDS: VGPR v holds K = v + 2*khalf
      const int jb = jstart + kk * 4 + khalf * 2;
      v2f b;
      b[0] = lds[jb * LDSLD + cw + M];
      b[1] = lds[(jb + 1) * LDSLD + cw + M];
      if (kk & 1)
        acc1 = __builtin_amdgcn_wmma_f32_16x16x4_f32(
            false, a, false, b, (short)0, acc1, false, false);
      else
        acc0 = __builtin_amdgcn_wmma_f32_16x16x4_f32(
            false, a, false, b, (short)0, acc0, false, false);
    }
    const v8f acc = acc0 + acc1;

    // D layout: VGPR v, lanes0-15 -> row v, lanes16-31 -> row v+8; col = M
    // Written once, never re-read: non-temporal stores.
    const int t0 = tt * 16;
    #pragma unroll
    for (int v = 0; v < 8; ++v) {
      __builtin_nontemporal_store(
          acc[v], out + (size_t)(t0 + v + 8 * khalf) * CCOLS + c0 + cw + M);
    }
  }
}

// ---------------------------------------------------------------------------
extern "C" void kernel_launch(void* const* d_in, const int* in_sizes, int n_in,
                              void* d_out, int out_size, void* d_ws, size_t ws_size,
                              hipStream_t stream) {
  const float* X     = (const float*)d_in[0];   // [128, 8192, 64] f32
  const float* M_pad = (const float*)d_in[1];   // [32, 32] f32
  const float* M_big = (const float*)d_in[2];   // [95, 32] f32
  float* out = (float*)d_out;                   // [128, 8192, 64] f32
  float* wsA = (float*)d_ws;                    // needs 8*12*32*2*4 = 24 KB

  band_weights_66675072303670_kernel<<<NTILE, 32, 0, stream>>>(M_pad, M_big, wsA);

  const int nblocks = CCOLS / 128;              // 4096
  band_mix_66675072303670_kernel<<<nblocks, 256, 0, stream>>>(X, wsA, out);
}